// _PureMamba_52802327937055
// MI455X (gfx1250) — compile-verified
//
#include <hip/hip_runtime.h>
#include <math.h>

// ---------------------------------------------------------------------------
// Mamba block forward for MI455X (gfx1250, wave32, WMMA + async-to-LDS).
//   D_MODEL=1024, D_STATE=16, D_CONV=4, D_INNER=2048, BATCH=2, L=2048
// Pipeline:
//   K1  xz   = x @ W_in                      (WMMA f32 16x16x4, 4096x4096, K=1024)
//   K2  xconv= silu(depthwise_conv(x_)+b)
//   K3  bcd  = xconv @ W_x                   (skinny, N=33)
//   K4  scan (16 states in regs, bcd rows staged via LDS)
//   K5  out  = yz @ W_out                    (WMMA f32 16x16x4, 4096x1024, K=2048)
// ---------------------------------------------------------------------------

typedef __attribute__((ext_vector_type(2))) float v2f;
typedef __attribute__((ext_vector_type(8))) float v8f;
typedef int gv4i __attribute__((vector_size(16)));   // matches builtin param type

#define TOKENS   4096
#define DMODEL   1024
#define DINNER   2048
#define DSTATE   16
#define SEQL     2048
#define NBCD     33

#define GLOBAL_AS __attribute__((address_space(1)))
#define LDS_AS    __attribute__((address_space(3)))

#if __has_builtin(__builtin_amdgcn_global_load_async_to_lds_b128) && \
    __has_builtin(__builtin_amdgcn_global_load_async_to_lds_b32)
#define HAVE_ASYNC_LDS 1
#endif

__device__ __forceinline__ void async_wait0() {
#if __has_builtin(__builtin_amdgcn_s_wait_asynccnt)
  __builtin_amdgcn_s_wait_asynccnt(0);
#else
  asm volatile("s_wait_asynccnt 0x0" ::: "memory");
#endif
}

// ----------------------------- WMMA f32 GEMM -------------------------------
// C[M,N] = A[M,K] @ B[K,N], all row-major fp32. M%32==0, N%64==0, K%32==0.
// Block: 256 threads (8 waves, 2x4 tile grid), block tile 32x64, K chunk 32.
__global__ __launch_bounds__(256)
void mamba_wmma_gemm_f32(const float* __restrict__ A,
                         const float* __restrict__ B,
                         float* __restrict__ C,
                         int M, int N, int K) {
  __shared__ float As[32][36];   // stride 36: 16B-aligned rows, 36*i distinct mod 64 (i<16)
  __shared__ float Bs[32][72];   // stride 72: half-wave K rows on disjoint banks

  const int tid   = threadIdx.x;
  const int lane  = tid & 31;
  const int wave  = tid >> 5;
  const int wm    = wave >> 2;        // 0..1
  const int wn    = wave & 3;         // 0..3
  const int lhalf = lane >> 4;        // 0,1
  const int l15   = lane & 15;
  const int m0    = blockIdx.y * 32;
  const int n0    = blockIdx.x * 64;

  const int ai = tid >> 3;            // A stage: row 0..31
  const int aj = (tid & 7) << 2;      // A stage: col 0,4,..28

  v8f acc = {0.f, 0.f, 0.f, 0.f, 0.f, 0.f, 0.f, 0.f};

  for (int k0 = 0; k0 < K; k0 += 32) {
#if defined(HAVE_ASYNC_LDS)
    // ---- async stage A tile 32x32 (one b128 per lane, memory -> LDS direct) ----
    __builtin_amdgcn_global_load_async_to_lds_b128(
        (GLOBAL_AS gv4i*)(A + (size_t)(m0 + ai) * K + (k0 + aj)),
        (LDS_AS gv4i*)&As[ai][aj], 0, 0);
    // ---- async stage B tile 32x64 (two b128 per lane) ----
    #pragma unroll
    for (int q = 0; q < 2; ++q) {
      const int idx = tid * 2 + q;        // 0..511 float4 slots
      const int r   = idx >> 4;           // row 0..31
      const int c4  = (idx & 15) << 2;    // col 0,4,..60
      __builtin_amdgcn_global_load_async_to_lds_b128(
          (GLOBAL_AS gv4i*)(B + (size_t)(k0 + r) * N + (n0 + c4)),
          (LDS_AS gv4i*)&Bs[r][c4], 0, 0);
    }
    if (k0 + 32 < K)  // prefetch next A tile -> global_prefetch_b8
      __builtin_prefetch(A + (size_t)(m0 + ai) * K + (k0 + 32 + aj), 0, 1);
    async_wait0();       // this wave's async transfers landed in LDS
    __syncthreads();     // all waves' transfers visible
#else
    {
      const float4 v = *(const float4*)(A + (size_t)(m0 + ai) * K + (k0 + aj));
      As[ai][aj + 0] = v.x; As[ai][aj + 1] = v.y;
      As[ai][aj + 2] = v.z; As[ai][aj + 3] = v.w;
      if (k0 + 32 < K)
        __builtin_prefetch(A + (size_t)(m0 + ai) * K + (k0 + 32 + aj), 0, 1);
    }
    #pragma unroll
    for (int q = 0; q < 2; ++q) {
      const int idx = tid * 2 + q;
      const int r   = idx >> 4;
      const int c4  = (idx & 15) << 2;
      const float4 v = *(const float4*)(B + (size_t)(k0 + r) * N + (n0 + c4));
      Bs[r][c4 + 0] = v.x; Bs[r][c4 + 1] = v.y;
      Bs[r][c4 + 2] = v.z; Bs[r][c4 + 3] = v.w;
    }
    __syncthreads();
#endif

    // ---- 8 WMMAs per chunk (K step 4) ----
    #pragma unroll
    for (int kk = 0; kk < 32; kk += 4) {
      const int kb = kk + (lhalf << 1);   // this lane's K base
      v2f a; a.x = As[wm * 16 + l15][kb];     a.y = As[wm * 16 + l15][kb + 1];
      v2f b; b.x = Bs[kb + 0][wn * 16 + l15]; b.y = Bs[kb + 1][wn * 16 + l15];
      acc = __builtin_amdgcn_wmma_f32_16x16x4_f32(
          /*neg_a=*/false, a, /*neg_b=*/false, b,
          /*c_mod=*/(short)0, acc, /*reuse_a=*/false, /*reuse_b=*/false);
    }
    __syncthreads();
  }

  // ---- store D: VGPR r -> row r + 8*lhalf ----
  #pragma unroll
  for (int r = 0; r < 8; ++r) {
    const int row = m0 + wm * 16 + r + 8 * lhalf;
    const int col = n0 + wn * 16 + l15;
    C[(size_t)row * N + col] = acc[r];
  }
}

// ------------------------ depthwise conv + SiLU ----------------------------
__global__ __launch_bounds__(256)
void mamba_conv_silu(const float* __restrict__ xz,      // [TOKENS][2*DINNER], x_ = cols 0..DINNER-1
                     const float* __restrict__ conv_w,  // [DINNER][4]
                     const float* __restrict__ conv_b,  // [DINNER]
                     float* __restrict__ xconv) {       // [TOKENS][DINNER]
  const int d     = blockIdx.x * 256 + threadIdx.x;
  const int token = blockIdx.y;
  const int l     = token & (SEQL - 1);

  float s = conv_b[d];
  #pragma unroll
  for (int j = 0; j < 4; ++j) {
    const int ll = l - 3 + j;
    if (ll >= 0)
      s += conv_w[d * 4 + j] * xz[(size_t)(token - 3 + j) * (2 * DINNER) + d];
  }
  const float v = s;
  xconv[(size_t)token * DINNER + d] = v / (1.f + __expf(-v));
}

// --------------------------- bcd = xconv @ W_x -----------------------------
__global__ __launch_bounds__(256)
void mamba_bcd(const float* __restrict__ xconv,  // [TOKENS][DINNER]
               const float* __restrict__ W_x,    // [DINNER][NBCD]
               float* __restrict__ bcd) {        // [TOKENS][NBCD]
  const int idx = blockIdx.x * 256 + threadIdx.x;
  if (idx >= TOKENS * NBCD) return;
  const int token = idx / NBCD;
  const int n     = idx - token * NBCD;
  const float* xr = xconv + (size_t)token * DINNER;
  float acc = 0.f;
  #pragma unroll 4
  for (int k = 0; k < DINNER; ++k)
    acc += xr[k] * W_x[(size_t)k * NBCD + n];
  bcd[idx] = acc;
}

// ------------------------------ selective scan -----------------------------
// grid (DINNER/256, BATCH); each thread owns one channel d, 16 states in regs.
// bcd rows (33 floats/step, shared by all channels) staged via LDS, 64 steps/chunk.
__global__ __launch_bounds__(256)
void mamba_scan(const float* __restrict__ bcd,     // [TOKENS][NBCD]
                const float* __restrict__ xconv,   // [TOKENS][DINNER]
                const float* __restrict__ xz,      // z = cols DINNER..2*DINNER-1
                const float* __restrict__ w_dt,
                const float* __restrict__ b_dt,
                const float* __restrict__ A_log,   // [DINNER][DSTATE]
                const float* __restrict__ Dp,
                float* __restrict__ yz) {          // [TOKENS][DINNER]
  __shared__ float sb[64][34];                     // 64 timesteps x 33 cols (padded)
  const int tid = threadIdx.x;
  const int d   = blockIdx.x * 256 + tid;
  const int b   = blockIdx.y;

  float a[DSTATE], h[DSTATE];
  #pragma unroll
  for (int n = 0; n < DSTATE; ++n) {
    a[n] = -__expf(A_log[(size_t)d * DSTATE + n]);
    h[n] = 0.f;
  }
  const float wdt = w_dt[d], bdt = b_dt[d], dpv = Dp[d];

  for (int t0 = 0; t0 < SEQL; t0 += 64) {
    __syncthreads();
#if defined(HAVE_ASYNC_LDS)
    for (int idx = tid; idx < 64 * NBCD; idx += 256) {
      const int r = idx / NBCD, c = idx - r * NBCD;
      __builtin_amdgcn_global_load_async_to_lds_b32(
          (GLOBAL_AS int*)(bcd + ((size_t)(b * SEQL + t0 + r)) * NBCD + c),
          (LDS_AS int*)&sb[r][c], 0, 0);
    }
    async_wait0();
#else
    for (int idx = tid; idx < 64 * NBCD; idx += 256) {
      const int r = idx / NBCD, c = idx - r * NBCD;
      sb[r][c] = bcd[((size_t)(b * SEQL + t0 + r)) * NBCD + c];
    }
#endif
    __syncthreads();

    for (int tt = 0; tt < 64; ++tt) {
      const int token = b * SEQL + t0 + tt;
      const float dtraw = sb[tt][32] * wdt + bdt;
      const float dt = (dtraw > 20.f) ? dtraw : log1pf(__expf(dtraw));  // softplus
      const float xv = xconv[(size_t)token * DINNER + d];
      const float dx = dt * xv;
      float y = 0.f;
      #pragma unroll
      for (int n = 0; n < DSTATE; ++n) {
        const float dA = __expf(dt * a[n]);
        h[n] = dA * h[n] + dx * sb[tt][n];        // B_t = cols 0..15
        y   += h[n] * sb[tt][DSTATE + n];          // C_t = cols 16..31
      }
      y += xv * dpv;
      const float zv = xz[(size_t)token * (2 * DINNER) + DINNER + d];
      yz[(size_t)token * DINNER + d] = y * (zv / (1.f + __expf(-zv)));
    }
  }
}

// ------------------------------- launcher ----------------------------------
extern "C" void kernel_launch(void* const* d_in, const int* in_sizes, int n_in,
                              void* d_out, int out_size, void* d_ws, size_t ws_size,
                              hipStream_t stream) {
  const float* x      = (const float*)d_in[0];
  const float* W_in   = (const float*)d_in[1];
  const float* conv_w = (const float*)d_in[2];
  const float* conv_b = (const float*)d_in[3];
  const float* W_x    = (const float*)d_in[4];
  const float* w_dt   = (const float*)d_in[5];
  const float* b_dt   = (const float*)d_in[6];
  const float* A_log  = (const float*)d_in[7];
  const float* Dp     = (const float*)d_in[8];
  const float* W_out  = (const float*)d_in[9];

  float* ws    = (float*)d_ws;
  float* xz    = ws;                                      // 4096*4096  (64 MB)
  float* xconv = xz    + (size_t)TOKENS * 2 * DINNER;     // 4096*2048  (32 MB)
  float* bcd   = xconv + (size_t)TOKENS * DINNER;         // 4096*33    (0.52 MB)
  float* yz    = bcd   + (size_t)TOKENS * NBCD;           // 4096*2048  (32 MB)

  // K1: xz = x @ W_in   (M=4096, N=4096, K=1024)
  mamba_wmma_gemm_f32<<<dim3((2 * DINNER) / 64, TOKENS / 32), 256, 0, stream>>>(
      x, W_in, xz, TOKENS, 2 * DINNER, DMODEL);

  // K2: depthwise conv + SiLU
  mamba_conv_silu<<<dim3(DINNER / 256, TOKENS), 256, 0, stream>>>(
      xz, conv_w, conv_b, xconv);

  // K3: bcd = xconv @ W_x
  mamba_bcd<<<(TOKENS * NBCD + 255) / 256, 256, 0, stream>>>(xconv, W_x, bcd);

  // K4: selective scan + fuse (+x*Dp) * silu(z)
  mamba_scan<<<dim3(DINNER / 256, 2), 256, 0, stream>>>(
      bcd, xconv, xz, w_dt, b_dt, A_log, Dp, yz);

  // K5: out = yz @ W_out  (M=4096, N=1024, K=2048)
  mamba_wmma_gemm_f32<<<dim3(DMODEL / 64, TOKENS / 32), 256, 0, stream>>>(
      yz, W_out, (float*)d_out, TOKENS, DMODEL, DINNER);
}